// SubsetOperator_28793460753037
// MI455X (gfx1250) — compile-verified
//
#include <hip/hip_runtime.h>

// -------------------------------------------------------------------------
// SubsetOperator (Gumbel top-K relaxed subset), B*S = 8192 rows, N = 4096.
// Algebraic transform: softmax(s0 + sum log m_i) == normalize(exp(s0-max) * prod m_i)
// -> entire K=16 loop runs on register-resident weights with ZERO transcendentals
//    in the loop (one v_exp_f32 per element total). Kernel is HBM-bound:
//    384 MiB / 23.3 TB/s ~= 17 us floor.
// CDNA5 path: rows staged via global_load_async_to_lds_b128 + s_wait_asynccnt.
// -------------------------------------------------------------------------

#define AS1 __attribute__((address_space(1)))
#define AS3 __attribute__((address_space(3)))

typedef int v4i __attribute__((ext_vector_type(4)));

#define N_COLS   4096
#define TPB      256
#define F4_PER_T 4          // 4096 floats / 256 threads / 4 per float4
#define KSEL     16
#define EPS_TINY 1.17549435e-38f

__device__ __forceinline__ void async_copy_f4(const float4* gsrc, float4* lds_dst) {
#if __has_builtin(__builtin_amdgcn_global_load_async_to_lds_b128)
  __builtin_amdgcn_global_load_async_to_lds_b128(
      (AS1 v4i*)gsrc, (AS3 v4i*)lds_dst, /*offset=*/0, /*cpol=*/0);
#else
  unsigned lds_off = (unsigned)(unsigned long long)(AS3 char*)(void*)lds_dst;
  asm volatile("global_load_async_to_lds_b128 %0, %1, off"
               :: "v"(lds_off), "v"(gsrc) : "memory");
#endif
}

__device__ __forceinline__ void wait_async_zero() {
#if __has_builtin(__builtin_amdgcn_s_wait_asynccnt)
  __builtin_amdgcn_s_wait_asynccnt(0);
#else
  asm volatile("s_wait_asynccnt 0" ::: "memory");
#endif
}

__global__ __launch_bounds__(TPB)
void subset_op_kernel(const float* __restrict__ scores,
                      const float* __restrict__ gnoise,
                      float* __restrict__ out) {
  __shared__ float lds_s[N_COLS];
  __shared__ float lds_g[N_COLS];
  __shared__ float red_max[8];
  __shared__ float red_sum[2][8];

  const int t    = threadIdx.x;
  const int lane = t & 31;
  const int wid  = t >> 5;

  const long long rowbase = (long long)blockIdx.x * (long long)N_COLS;
  const float4* gs = (const float4*)(scores + rowbase);
  const float4* gg = (const float4*)(gnoise + rowbase);
  float4* ls = (float4*)lds_s;
  float4* lg = (float4*)lds_g;

  // ---- async stage both rows into LDS (each lane later reads back only
  //      the bytes its own async loads wrote -> s_wait_asynccnt suffices) ----
#pragma unroll
  for (int j = 0; j < F4_PER_T; ++j) {
    const int v = t + j * TPB;
    async_copy_f4(gs + v, ls + v);
    async_copy_f4(gg + v, lg + v);
  }
  wait_async_zero();

  // ---- s0 = scores + g, find row max ----
  float w[KSEL];     // 16 perturbed scores -> weights (register resident)
  float kh[KSEL];    // khot accumulator
  float m = -3.402823466e38f;
#pragma unroll
  for (int j = 0; j < F4_PER_T; ++j) {
    const int v = t + j * TPB;
    float4 a = ls[v];
    float4 b = lg[v];
    float s0 = a.x + b.x, s1 = a.y + b.y, s2 = a.z + b.z, s3 = a.w + b.w;
    w[j * 4 + 0] = s0; w[j * 4 + 1] = s1; w[j * 4 + 2] = s2; w[j * 4 + 3] = s3;
    m = fmaxf(m, fmaxf(fmaxf(s0, s1), fmaxf(s2, s3)));
  }
  // wave32 max reduction
#pragma unroll
  for (int d = 16; d >= 1; d >>= 1)
    m = fmaxf(m, __shfl_xor(m, d, 32));
  if (lane == 0) red_max[wid] = m;
  __syncthreads();
  float M = red_max[0];
#pragma unroll
  for (int k = 1; k < 8; ++k) M = fmaxf(M, red_max[k]);

  // one exp per element (only transcendental in the whole kernel)
#pragma unroll
  for (int j = 0; j < KSEL; ++j) {
    w[j]  = __expf(w[j] - M);
    kh[j] = 0.0f;
  }

  // ---- K iterations: onehot = w / sum(w); khot += onehot;
  //      w <- onehot * max(1 - onehot, EPS)   (renormalized each iter) ----
  for (int it = 0; it < KSEL; ++it) {
    float ps = 0.0f;
#pragma unroll
    for (int j = 0; j < KSEL; ++j) ps += w[j];
#pragma unroll
    for (int d = 16; d >= 1; d >>= 1)
      ps += __shfl_xor(ps, d, 32);
    const int p = it & 1;
    if (lane == 0) red_sum[p][wid] = ps;
    __syncthreads();                 // double-buffered -> 1 barrier / iter
    float S = red_sum[p][0];
#pragma unroll
    for (int k = 1; k < 8; ++k) S += red_sum[p][k];
    const float inv = 1.0f / S;
#pragma unroll
    for (int j = 0; j < KSEL; ++j) {
      const float oh = w[j] * inv;                 // onehot_j
      kh[j] += oh;                                 // khot accumulate
      w[j]   = oh * fmaxf(1.0f - oh, EPS_TINY);    // masked + renormalized
    }
  }

  // ---- store khot ----
  float4* o4 = (float4*)(out + rowbase);
#pragma unroll
  for (int j = 0; j < F4_PER_T; ++j) {
    const int v = t + j * TPB;
    float4 r;
    r.x = kh[j * 4 + 0]; r.y = kh[j * 4 + 1];
    r.z = kh[j * 4 + 2]; r.w = kh[j * 4 + 3];
    o4[v] = r;
  }
}

extern "C" void kernel_launch(void* const* d_in, const int* in_sizes, int n_in,
                              void* d_out, int out_size, void* d_ws, size_t ws_size,
                              hipStream_t stream) {
  const float* scores = (const float*)d_in[0];
  const float* gnoise = (const float*)d_in[1];
  float* out = (float*)d_out;
  const int rows = in_sizes[0] / N_COLS;   // 4 * 2048 = 8192
  subset_op_kernel<<<dim3(rows), dim3(TPB), 0, stream>>>(scores, gnoise, out);
}